// EvoSABlock_65180423685194
// MI455X (gfx1250) — compile-verified
//
#include <hip/hip_runtime.h>
#include <math.h>

#define B_    16
#define C_    384
#define H_    28
#define W_    28
#define HW_   784
#define N_    785          // tokens = 1 + H*W
#define M_    (B_*N_)      // 12560 rows
#define MP_   (M_ + 16)    // padded rows so 32-row GEMM macro-tiles never read OOB
#define HEADS_ 8
#define D_    48
#define C3_   1152
#define C4_   1536
#define MT_   (M_/16)          // 785 16-row tiles
#define MB_   ((MT_ + 1)/2)    // 393 32-row macro-tiles
#define SLD   800              // padded score row stride (>= 788, 8B-aligned rows)

typedef __attribute__((ext_vector_type(2))) float v2f;
typedef __attribute__((ext_vector_type(8))) float v8f;

// D = A(16x4,f32) * B(4x16,f32) + C  — native fp32 matrix op on CDNA5
__device__ __forceinline__ v8f wmma4(v2f a, v2f b, v8f c) {
  return __builtin_amdgcn_wmma_f32_16x16x4_f32(false, a, false, b, (short)0, c,
                                               false, false);
}

// ---------------------------------------------------------------------------
// Depthwise 3x3 SAME conv + bias + residual, scattered into token layout
// ---------------------------------------------------------------------------
__global__ void conv_pe_kernel(const float* __restrict__ x,
                               const float* __restrict__ cw,
                               const float* __restrict__ cb,
                               float* __restrict__ xc) {
  int idx = blockIdx.x * 256 + threadIdx.x;
  if (idx >= B_ * C_ * HW_) return;
  int b   = idx / (C_ * HW_);
  int rem = idx % (C_ * HW_);
  int c   = rem / HW_;
  int p   = rem % HW_;
  int hh  = p / W_, ww = p % W_;
  const float* xp = x + (size_t)(b * C_ + c) * HW_;
  const float* wp = cw + c * 9;
  float s = 0.f;
#pragma unroll
  for (int i = 0; i < 3; ++i) {
    int hy = hh + i - 1;
    if (hy < 0 || hy >= H_) continue;
#pragma unroll
    for (int j = 0; j < 3; ++j) {
      int wx = ww + j - 1;
      if (wx < 0 || wx >= W_) continue;
      s += xp[hy * W_ + wx] * wp[i * 3 + j];
    }
  }
  xc[((size_t)b * N_ + 1 + p) * C_ + c] = xp[p] + s + cb[c];
}

__global__ void cls_copy_kernel(const float* __restrict__ cls,
                                float* __restrict__ xc) {
  int idx = blockIdx.x * 256 + threadIdx.x;
  if (idx >= B_ * C_) return;
  int b = idx / C_, c = idx % C_;
  xc[(size_t)b * N_ * C_ + c] = cls[idx];
}

// ---------------------------------------------------------------------------
// LayerNorm over C=384, one wave per token
// ---------------------------------------------------------------------------
__global__ void ln_kernel(const float* __restrict__ xin,
                          const float* __restrict__ g,
                          const float* __restrict__ bta,
                          float* __restrict__ yout, int ntok) {
  int wave = threadIdx.x >> 5, lane = threadIdx.x & 31;
  int t = blockIdx.x * 8 + wave;
  if (t >= ntok) return;
  const float* xp = xin + (size_t)t * C_;
  float s = 0.f, sq = 0.f;
#pragma unroll
  for (int i = 0; i < C_ / 32; ++i) {
    float v = xp[lane + i * 32];
    s += v; sq += v * v;
  }
#pragma unroll
  for (int off = 16; off; off >>= 1) {
    s  += __shfl_xor(s,  off, 32);
    sq += __shfl_xor(sq, off, 32);
  }
  float mu  = s  * (1.f / C_);
  float var = sq * (1.f / C_) - mu * mu;
  float rs  = rsqrtf(var + 1e-5f);
  float* yp = yout + (size_t)t * C_;
#pragma unroll
  for (int i = 0; i < C_ / 32; ++i) {
    int c = lane + i * 32;
    yp[c] = (xp[c] - mu) * rs * g[c] + bta[c];
  }
}

// ---------------------------------------------------------------------------
// Y = X(MxK) @ W(NxK)^T, register-blocked WMMA: wave = 32x64 macro-tile
// (2 m-tiles x 4 n-tiles, 8 accumulators; 6 b64 loads feed 8 WMMAs per K=4).
// EPI: 0 = store (+bias), 1 = Out += acc+bias (residual), 2 = exact GELU
// X buffers are MP_-row padded, so macro-tile reads never go OOB; stores are
// guarded per-row. Early exit is whole-wave, so EXEC is all-ones at WMMA.
// ---------------------------------------------------------------------------
template <int EPI>
__global__ __launch_bounds__(256) void gemm_wmma_kernel(
    const float* __restrict__ X, const float* __restrict__ Wt,
    const float* __restrict__ bias, float* __restrict__ Out,
    int K, int ldo, int Mblocks) {
  int lane = threadIdx.x & 31;
  int wave = threadIdx.x >> 5;
  int mblk = blockIdx.y * 8 + wave;
  if (mblk >= Mblocks) return;
  int n0 = blockIdx.x * 64;
  int m0 = mblk * 32;
  int rc = lane & 15;                          // A row == B col lane pattern
  int kb = (lane >> 4) << 1;                   // K pair base: 0 or 2
  const float* xp0 = X  + (size_t)(m0 + rc)      * K + kb;
  const float* xp1 = X  + (size_t)(m0 + 16 + rc) * K + kb;
  const float* wp0 = Wt + (size_t)(n0 + rc)      * K + kb;
  const float* wp1 = Wt + (size_t)(n0 + 16 + rc) * K + kb;
  const float* wp2 = Wt + (size_t)(n0 + 32 + rc) * K + kb;
  const float* wp3 = Wt + (size_t)(n0 + 48 + rc) * K + kb;

  v8f acc[2][4] = {};
#pragma unroll 4
  for (int k = 0; k < K; k += 4) {
    v2f a0 = *(const v2f*)(xp0 + k);
    v2f a1 = *(const v2f*)(xp1 + k);
    v2f b0 = *(const v2f*)(wp0 + k);
    v2f b1 = *(const v2f*)(wp1 + k);
    v2f b2 = *(const v2f*)(wp2 + k);
    v2f b3 = *(const v2f*)(wp3 + k);
    acc[0][0] = wmma4(a0, b0, acc[0][0]);
    acc[1][0] = wmma4(a1, b0, acc[1][0]);
    acc[0][1] = wmma4(a0, b1, acc[0][1]);
    acc[1][1] = wmma4(a1, b1, acc[1][1]);
    acc[0][2] = wmma4(a0, b2, acc[0][2]);
    acc[1][2] = wmma4(a1, b2, acc[1][2]);
    acc[0][3] = wmma4(a0, b3, acc[0][3]);
    acc[1][3] = wmma4(a1, b3, acc[1][3]);
  }

  int coln = lane & 15;
  int roff = (lane >> 4) << 3;
#pragma unroll
  for (int mi = 0; mi < 2; ++mi) {
#pragma unroll
    for (int j = 0; j < 4; ++j) {
      int nc = n0 + j * 16 + coln;
      float bv = bias ? bias[nc] : 0.f;
#pragma unroll
      for (int r = 0; r < 8; ++r) {
        int gm = m0 + mi * 16 + roff + r;
        if (gm < M_) {
          size_t oi = (size_t)gm * ldo + nc;
          float v = acc[mi][j][r] + bv;
          if (EPI == 0)      Out[oi] = v;
          else if (EPI == 1) Out[oi] += v;
          else               Out[oi] = 0.5f * v * (1.f + erff(v * 0.70710678118654752f));
        }
      }
    }
  }
}

// ---------------------------------------------------------------------------
// Fused attention: one wave per (b, h, 16-query-row tile).
// scores (16x785) in LDS -> softmax -> attn @ V, all via WMMA f32 16x16x4.
// ---------------------------------------------------------------------------
__global__ __launch_bounds__(32) void attn_kernel(const float* __restrict__ qkv,
                                                  float* __restrict__ o,
                                                  float* __restrict__ ga) {
  __shared__ float S[16 * SLD];               // 51.2 KB of the 320KB/WGP LDS
  int lane = threadIdx.x;
  int tile = blockIdx.x;                      // 0..49
  int h = blockIdx.y, b = blockIdx.z;
  int t0 = tile * 16;
  int row = lane & 15;
  int kb  = (lane >> 4) << 1;
  int gq = t0 + row; if (gq > N_ - 1) gq = N_ - 1;
  const float* qrow = qkv + (size_t)(b * N_ + gq) * C3_ + h * D_ + kb;
  const float scale = 0.14433756729740643f;   // 1/sqrt(48)

  // ---- scores: Q(16x48) @ K^T(48x785) in 50 column tiles (Q frags in regs)
  for (int ct = 0; ct < 50; ++ct) {
    int kc = ct * 16 + row; if (kc > N_ - 1) kc = N_ - 1;
    const float* krow = qkv + (size_t)(b * N_ + kc) * C3_ + C_ + h * D_ + kb;
    v8f acc = {};
#pragma unroll
    for (int k = 0; k < D_; k += 4) {
      v2f a  = *(const v2f*)(qrow + k);
      v2f bb = *(const v2f*)(krow + k);
      acc = wmma4(a, bb, acc);
    }
    int sc  = ct * 16 + (lane & 15);
    int mrb = (lane >> 4) << 3;
#pragma unroll
    for (int r = 0; r < 8; ++r) S[(mrb + r) * SLD + sc] = acc[r] * scale;
  }
  __syncthreads();

  // ---- softmax: lane pair per row, strided over 785 columns
  {
    int r = lane >> 1, sub = lane & 1;
    float mx = -3.0e38f;
    for (int c = sub; c < N_; c += 2) mx = fmaxf(mx, S[r * SLD + c]);
    mx = fmaxf(mx, __shfl_xor(mx, 1, 32));
    float sum = 0.f;
    for (int c = sub; c < N_; c += 2) {
      float e = expf(S[r * SLD + c] - mx);
      S[r * SLD + c] = e; sum += e;
    }
    sum += __shfl_xor(sum, 1, 32);
    float inv = 1.f / sum;
    for (int c = sub; c < N_; c += 2) S[r * SLD + c] *= inv;
    for (int c = N_ + sub; c < SLD; c += 2) S[r * SLD + c] = 0.f;  // zero pad cols
  }
  __syncthreads();

  // ---- cls-row attention probabilities for global_attn (tile 0, row 0)
  if (t0 == 0 && (lane >> 1) == 0) {
    int sub = lane & 1;
    float* gp = ga + (size_t)(b * HEADS_ + h) * HW_;
    for (int c = 1 + sub; c < N_; c += 2) gp[c - 1] = S[c];
  }

  // ---- attn(16x785, LDS) @ V(785x48): one A-frag feeds 3 n-tile WMMAs
  const float* vbase = qkv + (size_t)b * N_ * C3_ + 2 * C_ + h * D_;
  int coln = lane & 15;
  v8f acc0 = {}, acc1 = {}, acc2 = {};
  for (int k = 0; k < 788; k += 4) {
    int k0 = k + kb;
    v2f a = *(const v2f*)(&S[row * SLD + k0]);
    int r0 = k0 > HW_ ? HW_ : k0;              // clamp; padded S cols are zero
    int r1 = (k0 + 1) > HW_ ? HW_ : (k0 + 1);
    const float* vr0 = vbase + (size_t)r0 * C3_ + coln;
    const float* vr1 = vbase + (size_t)r1 * C3_ + coln;
    v2f b0, b1, b2;
    b0.x = vr0[0];  b0.y = vr1[0];
    b1.x = vr0[16]; b1.y = vr1[16];
    b2.x = vr0[32]; b2.y = vr1[32];
    acc0 = wmma4(a, b0, acc0);
    acc1 = wmma4(a, b1, acc1);
    acc2 = wmma4(a, b2, acc2);
  }
  int mrb = (lane >> 4) << 3;
#pragma unroll
  for (int r = 0; r < 8; ++r) {
    int gm = t0 + mrb + r;
    if (gm < N_) {
      float* op = o + ((size_t)(b * N_ + gm)) * C_ + h * D_ + coln;
      op[0]  = acc0[r];
      op[16] = acc1[r];
      op[32] = acc2[r];
    }
  }
}

// ---------------------------------------------------------------------------
// Assemble outputs: cls_out | x_out (token->NCHW transpose) | global_attn
// ---------------------------------------------------------------------------
__global__ void finalize_kernel(const float* __restrict__ xc,
                                const float* __restrict__ ga,
                                float* __restrict__ out) {
  size_t idx = (size_t)blockIdx.x * 256 + threadIdx.x;
  const size_t n_cls = (size_t)B_ * C_;
  const size_t n_x   = (size_t)B_ * C_ * HW_;
  const size_t n_ga  = (size_t)B_ * HW_;
  if (idx < n_cls) {
    int b = (int)(idx / C_), c = (int)(idx % C_);
    out[idx] = xc[(size_t)b * N_ * C_ + c];
  } else if (idx < n_cls + n_x) {
    size_t j = idx - n_cls;
    int b   = (int)(j / (C_ * HW_));
    int rem = (int)(j % (C_ * HW_));
    int c = rem / HW_, p = rem % HW_;
    out[idx] = xc[((size_t)b * N_ + 1 + p) * C_ + c];
  } else if (idx < n_cls + n_x + n_ga) {
    size_t j = idx - n_cls - n_x;
    int b = (int)(j / HW_), p = (int)(j % HW_);
    float s = 0.f;
#pragma unroll
    for (int hh = 0; hh < HEADS_; ++hh)
      s += ga[(size_t)(b * HEADS_ + hh) * HW_ + p];
    out[idx] = s * 0.125f;
  }
}

// ---------------------------------------------------------------------------
extern "C" void kernel_launch(void* const* d_in, const int* in_sizes, int n_in,
                              void* d_out, int out_size, void* d_ws, size_t ws_size,
                              hipStream_t stream) {
  const float* cls    = (const float*)d_in[0];
  const float* x      = (const float*)d_in[1];
  const float* conv_w = (const float*)d_in[2];
  const float* conv_b = (const float*)d_in[3];
  const float* ln1_g  = (const float*)d_in[4];
  const float* ln1_b  = (const float*)d_in[5];
  const float* qkv_w  = (const float*)d_in[6];
  const float* proj_w = (const float*)d_in[7];
  const float* proj_b = (const float*)d_in[8];
  const float* ln2_g  = (const float*)d_in[9];
  const float* ln2_b  = (const float*)d_in[10];
  const float* fc1_w  = (const float*)d_in[11];
  const float* fc1_b  = (const float*)d_in[12];
  const float* fc2_w  = (const float*)d_in[13];
  const float* fc2_b  = (const float*)d_in[14];

  // Workspace (all activation buffers MP_-row padded for 32-row macro-tiles):
  float* ws  = (float*)d_ws;
  float* xc  = ws;                           // MP*C   running residual
  float* y   = xc  + (size_t)MP_ * C_;       // MP*C   LN output (reused for LN2)
  float* qkv = y   + (size_t)MP_ * C_;       // MP*C3
  float* o   = qkv + (size_t)MP_ * C3_;      // MP*C   attention output
  float* m1  = qkv;                          // MP*C4  aliases dead qkv+o regions
  float* ga  = o   + (size_t)MP_ * C_;       // B*HEADS*HW cls-row probs
  float* out = (float*)d_out;

  conv_pe_kernel<<<(B_ * C_ * HW_ + 255) / 256, 256, 0, stream>>>(x, conv_w, conv_b, xc);
  cls_copy_kernel<<<(B_ * C_ + 255) / 256, 256, 0, stream>>>(cls, xc);

  ln_kernel<<<(M_ + 7) / 8, 256, 0, stream>>>(xc, ln1_g, ln1_b, y, M_);

  // qkv = y @ qkv_w^T        (12560 x 1152 x 384)
  gemm_wmma_kernel<0><<<dim3(C3_ / 64, (MB_ + 7) / 8), 256, 0, stream>>>(
      y, qkv_w, nullptr, qkv, C_, C3_, MB_);

  // fused softmax attention per (b,h,row-tile)
  attn_kernel<<<dim3(50, HEADS_, B_), 32, 0, stream>>>(qkv, o, ga);

  // xc += o @ proj_w^T + proj_b
  gemm_wmma_kernel<1><<<dim3(C_ / 64, (MB_ + 7) / 8), 256, 0, stream>>>(
      o, proj_w, proj_b, xc, C_, C_, MB_);

  ln_kernel<<<(M_ + 7) / 8, 256, 0, stream>>>(xc, ln2_g, ln2_b, y, M_);

  // m1 = gelu(y @ fc1_w^T + fc1_b)   (12560 x 1536 x 384)
  gemm_wmma_kernel<2><<<dim3(C4_ / 64, (MB_ + 7) / 8), 256, 0, stream>>>(
      y, fc1_w, fc1_b, m1, C_, C4_, MB_);

  // xc += m1 @ fc2_w^T + fc2_b      (12560 x 384 x 1536)
  gemm_wmma_kernel<1><<<dim3(C_ / 64, (MB_ + 7) / 8), 256, 0, stream>>>(
      m1, fc2_w, fc2_b, xc, C4_, C_, MB_);

  size_t tot = (size_t)B_ * C_ + (size_t)B_ * C_ * HW_ + (size_t)B_ * HW_;
  finalize_kernel<<<(unsigned)((tot + 255) / 256), 256, 0, stream>>>(xc, ga, out);
}